// KMeansQuantizer_65884798320944
// MI455X (gfx1250) — compile-verified
//
#include <hip/hip_runtime.h>
#include <hip/hip_bf16.h>

typedef float v2f __attribute__((ext_vector_type(2)));
typedef float v8f __attribute__((ext_vector_type(8)));

#define BS   16
#define LEN  2048
#define DIM  512
#define NG   2
#define VD   256
#define NC   320
#define GN_EPS 1e-5f

#define PROJ_TILES_PER_BG 256   // (2048/32) m-tiles * (256/64) n-tiles

// D = A(16x4 f32) * B(4x16 f32) + C(16x16 f32)   -> v_wmma_f32_16x16x4_f32
__device__ __forceinline__ v8f wmma_f32(v2f a, v2f b, v8f c) {
  return __builtin_amdgcn_wmma_f32_16x16x4_f32(false, a, false, b, (short)0, c,
                                               false, false);
}

// ---------------------------------------------------------------------------
// Projection: per (b,g) GEMM  ze[l,o] = sum_v x[b,l,g*256+v] * w[g,o,v]
// wave tile = 32x64, K=256, f32 WMMA 16x16x4 (8 WMMA per 6 loads per K-step).
// Deterministic GroupNorm partials: one fixed slot per wave tile.
// ---------------------------------------------------------------------------
__global__ void __launch_bounds__(256) k_proj(const float* __restrict__ x,
                                              const float* __restrict__ w,
                                              float* __restrict__ ze,
                                              float* __restrict__ partials) {
  const int lane = threadIdx.x & 31;
  const int wid  = blockIdx.x * (blockDim.x >> 5) + (threadIdx.x >> 5);
  const int bg = wid >> 8;          // 256 wave-tiles per (b,g)
  const int t  = wid & 255;
  const int mt = t >> 2, nt = t & 3;
  const int m0 = mt * 32, n0 = nt * 64;
  const int b = bg >> 1, g = bg & 1;

  const float* xrow = x + ((size_t)b * LEN) * DIM + g * VD;
  const float* wg   = w + (size_t)g * VD * VD;

  const int mrow = lane & 15;          // A: M = lane%16
  const int kb   = (lane >> 4) * 2;    // A/B: K pair base from lane>>4
  const int ncol = lane & 15;          // B: N = lane%16

  const float* aptr0 = xrow + (size_t)(m0 + mrow)      * DIM + kb;
  const float* aptr1 = xrow + (size_t)(m0 + mrow + 16) * DIM + kb;
  const float* bptr0 = wg + (size_t)(n0 + ncol)      * VD + kb;
  const float* bptr1 = wg + (size_t)(n0 + ncol + 16) * VD + kb;
  const float* bptr2 = wg + (size_t)(n0 + ncol + 32) * VD + kb;
  const float* bptr3 = wg + (size_t)(n0 + ncol + 48) * VD + kb;

  v8f acc[2][4] = {{v8f{}, v8f{}, v8f{}, v8f{}}, {v8f{}, v8f{}, v8f{}, v8f{}}};
#pragma unroll 8
  for (int k = 0; k < VD; k += 4) {
    v2f a0 = *(const v2f*)(aptr0 + k);
    v2f a1 = *(const v2f*)(aptr1 + k);
    v2f b0 = *(const v2f*)(bptr0 + k);
    v2f b1 = *(const v2f*)(bptr1 + k);
    v2f b2 = *(const v2f*)(bptr2 + k);
    v2f b3 = *(const v2f*)(bptr3 + k);
    acc[0][0] = wmma_f32(a0, b0, acc[0][0]);
    acc[0][1] = wmma_f32(a0, b1, acc[0][1]);
    acc[0][2] = wmma_f32(a0, b2, acc[0][2]);
    acc[0][3] = wmma_f32(a0, b3, acc[0][3]);
    acc[1][0] = wmma_f32(a1, b0, acc[1][0]);
    acc[1][1] = wmma_f32(a1, b1, acc[1][1]);
    acc[1][2] = wmma_f32(a1, b2, acc[1][2]);
    acc[1][3] = wmma_f32(a1, b3, acc[1][3]);
  }

  // store + partial GroupNorm stats. C layout: VGPR i -> rows i / i+8.
  float s = 0.f, ss = 0.f;
  const int rowoff = (lane >> 4) * 8;
  float* zbase = ze + ((size_t)bg * LEN) * VD;
#pragma unroll
  for (int mi = 0; mi < 2; ++mi) {
#pragma unroll
    for (int ni = 0; ni < 4; ++ni) {
#pragma unroll
      for (int i = 0; i < 8; ++i) {
        float v = acc[mi][ni][i];
        int row = m0 + mi * 16 + rowoff + i;
        int col = n0 + ni * 16 + (lane & 15);
        zbase[(size_t)row * VD + col] = v;
        s += v;
        ss += v * v;
      }
    }
  }

#pragma unroll
  for (int off = 16; off > 0; off >>= 1) {
    s  += __shfl_xor(s, off, 32);
    ss += __shfl_xor(ss, off, 32);
  }
  if (lane == 0) {
    partials[(size_t)bg * PROJ_TILES_PER_BG + t] = s;
    partials[(size_t)(32 + bg) * PROJ_TILES_PER_BG + t] = ss;
  }
}

// ---------------------------------------------------------------------------
// Deterministic finalize: mean/rstd per (b,g) from fixed-order partial sums;
// e2[c] = sum_v emb[c,v]^2
// ---------------------------------------------------------------------------
__global__ void k_stats(const float* __restrict__ partials,
                        const float* __restrict__ emb,
                        float* __restrict__ meanArr,
                        float* __restrict__ rstdArr,
                        float* __restrict__ e2) {
  int tid = threadIdx.x;
  if (tid < 32) {
    const float* ps = partials + (size_t)tid * PROJ_TILES_PER_BG;
    const float* pq = partials + (size_t)(32 + tid) * PROJ_TILES_PER_BG;
    float s = 0.f, ss = 0.f;
    for (int i = 0; i < PROJ_TILES_PER_BG; ++i) {  // fixed order -> deterministic
      s  += ps[i];
      ss += pq[i];
    }
    const float n = (float)(VD * LEN);
    float mean = s / n;
    float var  = ss / n - mean * mean;
    meanArr[tid] = mean;
    rstdArr[tid] = rsqrtf(var + GN_EPS);
  } else if (tid < 32 + NC) {
    int c = tid - 32;
    const float* e = emb + (size_t)c * VD;
    float s = 0.f;
#pragma unroll 8
    for (int v = 0; v < VD; ++v) s += e[v] * e[v];
    e2[c] = s;
  }
}

// ---------------------------------------------------------------------------
// VQ: normalize ze on the fly, WMMA dots vs 320 codes in 64-code chunks,
// argmin (first-index ties) per row, gather embedding row to out.
// wave tile = 16 rows.
// ---------------------------------------------------------------------------
__global__ void __launch_bounds__(256) k_vq(const float* __restrict__ ze,
                                            const float* __restrict__ gnw,
                                            const float* __restrict__ gnb,
                                            const float* __restrict__ emb,
                                            const float* __restrict__ meanArr,
                                            const float* __restrict__ rstdArr,
                                            const float* __restrict__ e2,
                                            float* __restrict__ out) {
  const int lane = threadIdx.x & 31;
  const int wid  = blockIdx.x * (blockDim.x >> 5) + (threadIdx.x >> 5);
  const int bg = wid >> 7;          // 128 row-tiles per (b,g)
  const int mt = wid & 127;
  const int l0 = mt * 16;
  const int b = bg >> 1, g = bg & 1;

  const float mean = meanArr[bg];
  const float rstd = rstdArr[bg];

  const int mrow = lane & 15;
  const int kb   = (lane >> 4) * 2;
  const int ncol = lane & 15;

  const float* zrow = ze + ((size_t)bg * LEN + l0 + mrow) * VD + kb;
  const float* gw   = gnw + g * VD + kb;
  const float* gb   = gnb + g * VD + kb;

  float bestv[8];
  int   besti[8];
#pragma unroll
  for (int i = 0; i < 8; ++i) { bestv[i] = 3.4e38f; besti[i] = 0; }

  for (int c0 = 0; c0 < NC; c0 += 64) {
    v8f acc[4] = {v8f{}, v8f{}, v8f{}, v8f{}};
#pragma unroll 4
    for (int k = 0; k < VD; k += 4) {
      v2f za = *(const v2f*)(zrow + k);
      v2f wv = *(const v2f*)(gw + k);
      v2f bv = *(const v2f*)(gb + k);
      v2f a;
      a.x = (za.x - mean) * rstd * wv.x + bv.x;
      a.y = (za.y - mean) * rstd * wv.y + bv.y;
#pragma unroll
      for (int nt = 0; nt < 4; ++nt) {
        const float* ep = emb + (size_t)(c0 + nt * 16 + ncol) * VD + kb + k;
        v2f bf = *(const v2f*)ep;
        acc[nt] = wmma_f32(a, bf, acc[nt]);
      }
    }
    // dist = e2[c] - 2*dot ; butterfly argmin across the 16-lane half-wave
#pragma unroll
    for (int nt = 0; nt < 4; ++nt) {
      int cbase = c0 + nt * 16 + ncol;
      float e2v = e2[cbase];
#pragma unroll
      for (int i = 0; i < 8; ++i) {
        float d = e2v - 2.0f * acc[nt][i];
        int   ci = cbase;
#pragma unroll
        for (int off = 1; off < 16; off <<= 1) {
          float od = __shfl_xor(d, off, 32);
          int   oi = __shfl_xor(ci, off, 32);
          if (od < d || (od == d && oi < ci)) { d = od; ci = oi; }
        }
        if (d < bestv[i] || (d == bestv[i] && ci < besti[i])) {
          bestv[i] = d;
          besti[i] = ci;
        }
      }
    }
  }

  // gather: out[b, l0+r, g*256 + :] = emb[best_r, :]
  float* obase = out + ((size_t)b * LEN + l0) * DIM + g * VD;
#pragma unroll
  for (int r = 0; r < 16; ++r) {
    int ci = __shfl(besti[r & 7], (r >> 3) * 16, 32);
    const float4* ep = (const float4*)(emb + (size_t)ci * VD);
    float4* op = (float4*)(obase + (size_t)r * DIM);
    op[lane]      = ep[lane];
    op[lane + 32] = ep[lane + 32];
  }
}

extern "C" void kernel_launch(void* const* d_in, const int* in_sizes, int n_in,
                              void* d_out, int out_size, void* d_ws, size_t ws_size,
                              hipStream_t stream) {
  (void)in_sizes; (void)n_in; (void)out_size; (void)ws_size;
  const float* x   = (const float*)d_in[0];
  const float* pw  = (const float*)d_in[1];
  const float* gnw = (const float*)d_in[2];
  const float* gnb = (const float*)d_in[3];
  const float* emb = (const float*)d_in[4];
  float* out = (float*)d_out;

  float* ws = (float*)d_ws;
  const size_t ZE_ELEMS = (size_t)BS * NG * LEN * VD;   // 16.7M floats = 64 MB
  float* ze       = ws;
  float* partials = ws + ZE_ELEMS;                       // [2*32*256] sum|sumsq
  float* meanArr  = partials + 2 * 32 * PROJ_TILES_PER_BG;  // [32]
  float* rstdArr  = meanArr + 32;                        // [32]
  float* e2       = rstdArr + 32;                        // [320]

  // 256 wave-tiles per (b,g), 8 waves per block -> 32 blocks per (b,g)
  hipLaunchKernelGGL(k_proj,  dim3(BS * NG * 32), dim3(256), 0, stream, x, pw, ze,
                     partials);
  hipLaunchKernelGGL(k_stats, dim3(1),            dim3(352), 0, stream, partials, emb,
                     meanArr, rstdArr, e2);
  hipLaunchKernelGGL(k_vq,    dim3(BS * NG * 16), dim3(256), 0, stream, ze, gnw, gnb,
                     emb, meanArr, rstdArr, e2, out);
}